// SpixAttentionv2_70841190580636
// MI455X (gfx1250) — compile-verified
//
#include <hip/hip_runtime.h>
#include <hip/hip_bf16.h>

typedef __attribute__((ext_vector_type(2))) float v2f;
typedef __attribute__((ext_vector_type(8))) float v8f;

#define BATCH 2
#define CDIM  256
#define HDIM  128
#define WDIM  128
#define NPIX  4096            // 64*64 downsampled pixels
#define NSEG  256             // S = n_h*n_w
#define NSEG1 257
#define E9    (9*NPIX)        // 36864 entries per batch
#define TAUC  0.07f
#define EPSF  1e-16f

// workspace float offsets
#define OFF_V    0
#define SZ_V     (BATCH*NPIX*CDIM)
#define OFF_KD   (OFF_V + SZ_V)
#define SZ_KD    (BATCH*NPIX*CDIM)
#define OFF_NQ   (OFF_KD + SZ_KD)
#define SZ_NQ    (BATCH*NSEG*CDIM)
#define OFF_AT   (OFF_NQ + SZ_NQ)
#define SZ_AT    (BATCH*E9)
#define OFF_SUM  (OFF_AT + SZ_AT)

__device__ __forceinline__ float wave_reduce_add(float v) {
    #pragma unroll
    for (int off = 16; off > 0; off >>= 1)
        v += __shfl_xor(v, off, 32);
    return v;
}

// ---------------- init: zero out_query/out_cls and segment sums ----------------
__global__ void k_init(float* __restrict__ outqc, float* __restrict__ gsums) {
    int t = blockIdx.x * blockDim.x + threadIdx.x;
    if (t < 2 * BATCH * NSEG * CDIM) outqc[t] = 0.0f;
    if (t < BATCH * NSEG1) gsums[t] = 0.0f;
}

// ---------------- normalize q vectors: (B,S,C) -> nq (B,S,C) ----------------
__global__ void k_qnorm(const float* __restrict__ q, float* __restrict__ nq) {
    int wid  = (blockIdx.x * blockDim.x + threadIdx.x) >> 5;  // 0..511
    int lane = threadIdx.x & 31;
    const float* src = q + (size_t)wid * CDIM;
    float vals[8];
    float ns = 0.0f;
    #pragma unroll
    for (int j = 0; j < 8; ++j) {
        float x = src[lane + 32 * j];
        vals[j] = x; ns += x * x;
    }
    ns = wave_reduce_add(ns);
    float scale = 1.0f / fmaxf(sqrtf(ns), 1e-12f);
    float* dst = nq + (size_t)wid * CDIM;
    #pragma unroll
    for (int j = 0; j < 8; ++j) dst[lane + 32 * j] = vals[j] * scale;
}

// ---------------- v projection GEMM: v[b,p,o] = sum_c vf[b,c,p]*vw[o,c] + vb[o] ----
// One wave computes a 16x64 strip: 4 accumulators share each A fragment per K-step,
// K-loop of V_WMMA_F32_16X16X4_F32 (fp32 in/out, matches reference precision).
__global__ void k_vproj(const float* __restrict__ vfeat, const float* __restrict__ vw,
                        const float* __restrict__ vb, float* __restrict__ vout) {
    int wave = (blockIdx.x * blockDim.x + threadIdx.x) >> 5;  // 0..2047
    int lane = threadIdx.x & 31;
    int ng = wave & 3;                 // N-group: columns ng*64 .. ng*64+63
    int mt = (wave >> 2) & 255;        // M-tile: rows mt*16 .. mt*16+15
    int b  = wave >> 10;
    int p0 = mt * 16;
    int n0 = ng * 64;
    int half = lane >> 4;              // 0: K pair {0,1}, 1: K pair {2,3}
    int mn   = lane & 15;              // A row / B col / D col

    v8f acc0 = {}, acc1 = {}, acc2 = {}, acc3 = {};
    const float* Abase = vfeat + (size_t)b * CDIM * NPIX + p0 + mn;  // A[m,k]=vf[b,k,p0+m]
    const float* B0 = vw + (size_t)(n0 +  0 + mn) * CDIM;            // B[k,n]=vw[n,k]
    const float* B1 = vw + (size_t)(n0 + 16 + mn) * CDIM;
    const float* B2 = vw + (size_t)(n0 + 32 + mn) * CDIM;
    const float* B3 = vw + (size_t)(n0 + 48 + mn) * CDIM;

    for (int k0 = 0; k0 < CDIM; k0 += 4) {
        int kk = k0 + 2 * half;
        v2f a, bm0, bm1, bm2, bm3;
        a.x   = Abase[(size_t)kk * NPIX];
        a.y   = Abase[(size_t)(kk + 1) * NPIX];
        bm0.x = B0[kk]; bm0.y = B0[kk + 1];
        bm1.x = B1[kk]; bm1.y = B1[kk + 1];
        bm2.x = B2[kk]; bm2.y = B2[kk + 1];
        bm3.x = B3[kk]; bm3.y = B3[kk + 1];
        acc0 = __builtin_amdgcn_wmma_f32_16x16x4_f32(false, a, false, bm0, (short)0, acc0, false, false);
        acc1 = __builtin_amdgcn_wmma_f32_16x16x4_f32(false, a, false, bm1, (short)0, acc1, false, false);
        acc2 = __builtin_amdgcn_wmma_f32_16x16x4_f32(false, a, false, bm2, (short)0, acc2, false, false);
        acc3 = __builtin_amdgcn_wmma_f32_16x16x4_f32(false, a, false, bm3, (short)0, acc3, false, false);
    }

    float* out = vout + (size_t)(b * NPIX + p0) * CDIM + n0 + mn;
    float bias0 = vb[n0 +  0 + mn];
    float bias1 = vb[n0 + 16 + mn];
    float bias2 = vb[n0 + 32 + mn];
    float bias3 = vb[n0 + 48 + mn];
    #pragma unroll
    for (int r = 0; r < 8; ++r) {
        int row = r + 8 * half;  // D: vgpr r -> M=r (lanes 0-15), M=r+8 (lanes 16-31)
        float* orow = out + (size_t)row * CDIM;
        orow[ 0] = acc0[r] + bias0;
        orow[16] = acc1[r] + bias1;
        orow[32] = acc2[r] + bias2;
        orow[48] = acc3[r] + bias3;
    }
}

// ---------------- k 2x2 downsample: k_down[b,p,c] (channel-last) ----------------
__global__ void k_kdown(const float* __restrict__ kf, float* __restrict__ kd) {
    int t = blockIdx.x * blockDim.x + threadIdx.x;  // B*C*NPIX
    int p = t & (NPIX - 1);
    int c = (t >> 12) & (CDIM - 1);
    int b = t >> 20;
    int h2 = p >> 6, w2 = p & 63;
    const float* r0 = kf + (((size_t)b * CDIM + c) * HDIM + 2 * h2) * WDIM + 2 * w2;
    float avg = 0.25f * (r0[0] + r0[1] + r0[WDIM] + r0[WDIM + 1]);
    kd[(size_t)(b * NPIX + p) * CDIM + c] = avg;
}

// ---------------- affinity: one block per (b, superpixel cell), 8x8 pixel tile ----
__global__ void k_affinity(const float* __restrict__ kf, const float* __restrict__ nq,
                           float* __restrict__ affout) {
    __shared__ float sQ[9][CDIM];
    int blk  = blockIdx.x;       // b*256 + cell
    int cell = blk & 255;
    int b    = blk >> 8;
    int cr = cell >> 4, cc = cell & 15;
    int tid = threadIdx.x;       // 256 threads

    #pragma unroll
    for (int n = 0; n < 9; ++n) {
        int nr = cr + n / 3 - 1, nc = cc + n % 3 - 1;
        bool inr = (nr >= 0 && nr < 16 && nc >= 0 && nc < 16);
        sQ[n][tid] = inr ? nq[((size_t)b * NSEG + nr * 16 + nc) * CDIM + tid] : 0.0f;
    }
    __syncthreads();

    int wv = tid >> 5, lane = tid & 31;
    for (int i = 0; i < 8; ++i) {
        int pidx = wv + 8 * i;          // 0..63 within tile
        int py = pidx >> 3, px = pidx & 7;
        int h = cr * 8 + py, w = cc * 8 + px;
        const float* kcol = kf + (size_t)b * CDIM * HDIM * WDIM + h * WDIM + w;
        float kv[8];
        float ns = 0.0f;
        #pragma unroll
        for (int j = 0; j < 8; ++j) {
            float x = kcol[(size_t)(lane + 32 * j) * (HDIM * WDIM)];
            kv[j] = x; ns += x * x;
        }
        ns = wave_reduce_add(ns);
        float scale = 1.0f / fmaxf(sqrtf(ns), 1e-12f);
        #pragma unroll
        for (int n = 0; n < 9; ++n) {
            int nr = cr + n / 3 - 1, nc = cc + n % 3 - 1;
            float d = 0.0f;
            #pragma unroll
            for (int j = 0; j < 8; ++j) d += kv[j] * sQ[n][lane + 32 * j];
            d = wave_reduce_add(d);
            if (lane == 0) {
                float val = (nr < 0 || nr > 15 || nc < 0 || nc > 15)
                          ? -__builtin_huge_valf()
                          : (d * scale) / TAUC;
                affout[((size_t)(b * 9 + n) * HDIM + h) * WDIM + w] = val;
            }
        }
    }
}

// ---------------- downsample affinity + softmax over 9 + segment sums ----------------
__global__ void k_attn(const float* __restrict__ aff, const int* __restrict__ spix,
                       float* __restrict__ attnc, float* __restrict__ gsums) {
    __shared__ float sSum[NSEG1];
    int b    = blockIdx.x >> 4;
    int pix0 = (blockIdx.x & 15) * 256;
    int tid  = threadIdx.x;
    for (int i = tid; i < NSEG1; i += 256) sSum[i] = 0.0f;
    __syncthreads();

    int pix = pix0 + tid;
    int h2 = pix >> 6, w2 = pix & 63;
    float e[9];
    float se = 0.0f;
    #pragma unroll
    for (int n = 0; n < 9; ++n) {
        const float* base = aff + ((size_t)(b * 9 + n) * HDIM + 2 * h2) * WDIM + 2 * w2;
        float a = 0.25f * (base[0] + base[1] + base[WDIM] + base[WDIM + 1]);
        e[n] = expf(a);              // exp(-inf) = 0 handles the boundary mask
        se += e[n];
    }
    float inv = 1.0f / (se + EPSF);
    #pragma unroll
    for (int n = 0; n < 9; ++n) {
        float ac = e[n] * inv;
        size_t eidx = (size_t)(b * 9 + n) * NPIX + pix;
        attnc[eidx] = ac;
        int s = spix[eidx];
        if (s < NSEG) atomicAdd(&sSum[s], ac);  // segment 256 stays zero (ref zeroes it)
    }
    __syncthreads();
    for (int i = tid; i < NSEG1; i += 256)
        atomicAdd(&gsums[b * NSEG1 + i], sSum[i]);
}

// ---------------- weighted scatter into out_query / out_cls ----------------
__global__ void k_scatter(const float* __restrict__ attnc, const float* __restrict__ gsums,
                          const int* __restrict__ spix, const float* __restrict__ vws,
                          const float* __restrict__ kdws, float* __restrict__ outq,
                          float* __restrict__ outc) {
    int wave = (blockIdx.x * blockDim.x + threadIdx.x) >> 5;  // 0..73727
    int lane = threadIdx.x & 31;
    int b = wave / E9;
    int r = wave - b * E9;
    int pix = r & (NPIX - 1);

    float ac    = attnc[(size_t)b * E9 + r];
    int   s0    = spix[r];                       // reference uses batch-0 indices for denom
    float denom = gsums[b * NSEG1 + s0];
    float wp    = ac / (denom + EPSF);
    int   s     = spix[(size_t)b * E9 + r];
    if (s >= NSEG) return;                       // out[:, :S] drops segment 256

    const float* vrow = vws  + (size_t)(b * NPIX + pix) * CDIM;
    const float* krow = kdws + (size_t)(b * NPIX + pix) * CDIM;
    float* oc = outc + (size_t)(b * NSEG + s) * CDIM;
    float* oq = outq + (size_t)(b * NSEG + s) * CDIM;
    #pragma unroll
    for (int j = 0; j < 8; ++j) {
        int c = lane + 32 * j;
        atomicAdd(&oc[c], wp * vrow[c]);
        atomicAdd(&oq[c], wp * krow[c]);
    }
}

extern "C" void kernel_launch(void* const* d_in, const int* in_sizes, int n_in,
                              void* d_out, int out_size, void* d_ws, size_t ws_size,
                              hipStream_t stream) {
    const float* q    = (const float*)d_in[0];
    const float* kf   = (const float*)d_in[1];
    const float* vf   = (const float*)d_in[2];
    const float* vw   = (const float*)d_in[3];
    const float* vb   = (const float*)d_in[4];
    const int*   spix = (const int*)d_in[5];

    float* out  = (float*)d_out;
    float* outq = out;                                   // (B,S,C)  out_query
    float* outc = out + BATCH * NSEG * CDIM;             // (B,S,C)  out_cls
    float* affo = out + 2 * BATCH * NSEG * CDIM;         // (B,9,H,W) affinity

    float* ws    = (float*)d_ws;
    float* v_ws  = ws + OFF_V;
    float* kd_ws = ws + OFF_KD;
    float* nq_ws = ws + OFF_NQ;
    float* at_ws = ws + OFF_AT;
    float* sm_ws = ws + OFF_SUM;

    k_init    <<<1024, 256, 0, stream>>>(outq, sm_ws);
    k_qnorm   <<<64, 256, 0, stream>>>(q, nq_ws);
    k_vproj   <<<256, 256, 0, stream>>>(vf, vw, vb, v_ws);   // 2048 waves, 16x64 strip each
    k_kdown   <<<(BATCH * CDIM * NPIX) / 256, 256, 0, stream>>>(kf, kd_ws);
    k_affinity<<<BATCH * 256, 256, 0, stream>>>(kf, nq_ws, affo);
    k_attn    <<<BATCH * 16, 256, 0, stream>>>(affo, spix, at_ws, sm_ws);
    k_scatter <<<(BATCH * E9) / 8, 256, 0, stream>>>(at_ws, sm_ws, spix, v_ws, kd_ws, outq, outc);
}